// GCNEncoder_54571854463791
// MI455X (gfx1250) — compile-verified
//
#include <hip/hip_runtime.h>
#include <cstdint>

// ---------------------------------------------------------------------------
// GCN (2x GCNConv, D=128) for MI455X / gfx1250.
//   N = 50000 nodes, E = 800000 edges, D = 128 features.
// Phase costs: GEMMs ~3.3 GFLOP (negligible); edge gather/scatter ~1.6 GB of
// L2-resident traffic (entire working set < 192MB L2) -> L2-atomic bound.
// GEMM uses V_WMMA_F32_16X16X4_F32 (exact fp32). Scatter uses 1 wave/edge:
// coalesced 512B float4 gather + hardware f32 global atomics.
// ---------------------------------------------------------------------------

#define DFEAT 128

typedef __attribute__((ext_vector_type(2))) float v2f;
typedef __attribute__((ext_vector_type(8))) float v8f;

// -------------------------------- utility ----------------------------------

__global__ void zero_f32(float* __restrict__ p, long n) {
  long i = (long)blockIdx.x * blockDim.x + threadIdx.x;
  long stride = (long)gridDim.x * blockDim.x;
  for (; i < n; i += stride) p[i] = 0.0f;
}

// deg[dst] += 1 for every edge (dst = second row of edge_index, int64)
__global__ void degree_count(const long long* __restrict__ dst, float* __restrict__ deg,
                             int nedges) {
  int i = blockIdx.x * blockDim.x + threadIdx.x;
  if (i < nedges) {
    int d = (int)dst[i];
    unsafeAtomicAdd(&deg[d], 1.0f);
  }
}

// in-place: deg -> rsqrt(deg + 1)   (self-loop included)
__global__ void rsqrt_inplace(float* __restrict__ deg, int n) {
  int i = blockIdx.x * blockDim.x + threadIdx.x;
  if (i < n) deg[i] = rsqrtf(deg[i] + 1.0f);
}

// ------------------------------- WMMA GEMM ---------------------------------
// H[N,128] = X[N,128] @ W[128,128]
// Block: 256 threads = 8 waves. Wave w owns N-tile w (cols 16w..16w+15) and
// M-tiles blockIdx.x*64 .. +63 (4 accumulators, B fragment reused 4x).
// Fragment layouts per CDNA5 ISA 7.12.2 (32-bit operands, wave32):
//   A 16x4 : lanes 0-15 rows, VGPR0=K, VGPR1=K+1; lanes 16-31 same rows, K+2/K+3
//   B 4x16 : lanes 0-15 cols, VGPR0=K, VGPR1=K+1; lanes 16-31 same cols, K+2/K+3
//   C 16x16: VGPR r -> row r (lanes 0-15, col=lane) / row r+8 (lanes 16-31)
__global__ void gcn_gemm_wmma(const float* __restrict__ X, const float* __restrict__ W,
                              float* __restrict__ H, int nrows) {
  const int wave  = threadIdx.x >> 5;        // 0..7 -> N-tile
  const int lane  = threadIdx.x & 31;
  const int l16   = lane & 15;
  const int hi    = lane >> 4;               // half-wave: 0 or 1
  const int nbase = wave * 16;
  const int mblk  = blockIdx.x * 64;

  v8f acc0 = {}, acc1 = {}, acc2 = {}, acc3 = {};

  const int r0 = mblk + l16;
  // clamp load rows (WMMA needs EXEC all-ones; OOB rows are masked at store)
  const long c0 = (long)((r0      < nrows) ? r0      : 0) * DFEAT;
  const long c1 = (long)((r0 + 16 < nrows) ? r0 + 16 : 0) * DFEAT;
  const long c2 = (long)((r0 + 32 < nrows) ? r0 + 32 : 0) * DFEAT;
  const long c3 = (long)((r0 + 48 < nrows) ? r0 + 48 : 0) * DFEAT;

  for (int k = 0; k < DFEAT; k += 4) {
    const int kk = k + hi * 2;
    v2f b;
    b.x = W[(long)kk * DFEAT + nbase + l16];
    b.y = W[(long)(kk + 1) * DFEAT + nbase + l16];

    v2f a0, a1, a2, a3;
    a0.x = X[c0 + kk]; a0.y = X[c0 + kk + 1];
    a1.x = X[c1 + kk]; a1.y = X[c1 + kk + 1];
    a2.x = X[c2 + kk]; a2.y = X[c2 + kk + 1];
    a3.x = X[c3 + kk]; a3.y = X[c3 + kk + 1];

    acc0 = __builtin_amdgcn_wmma_f32_16x16x4_f32(false, a0, false, b, (short)0, acc0, false, false);
    acc1 = __builtin_amdgcn_wmma_f32_16x16x4_f32(false, a1, false, b, (short)0, acc1, false, false);
    acc2 = __builtin_amdgcn_wmma_f32_16x16x4_f32(false, a2, false, b, (short)0, acc2, false, false);
    acc3 = __builtin_amdgcn_wmma_f32_16x16x4_f32(false, a3, false, b, (short)0, acc3, false, false);
  }

  const int col   = nbase + l16;
  const int rbase = mblk + hi * 8;
#pragma unroll
  for (int r = 0; r < 8; ++r) {
    int row = rbase + r;
    if (row < nrows)      H[(long)row * DFEAT + col] = acc0[r];
    if (row + 16 < nrows) H[(long)(row + 16) * DFEAT + col] = acc1[r];
    if (row + 32 < nrows) H[(long)(row + 32) * DFEAT + col] = acc2[r];
    if (row + 48 < nrows) H[(long)(row + 48) * DFEAT + col] = acc3[r];
  }
}

// ------------------------------ edge scatter -------------------------------
// One wave per edge: lane l handles features [4l, 4l+4).
//   gather: 32 lanes x float4 = one coalesced 512B row read (L2 resident)
//   scatter: hardware global_atomic_add_f32 into AGG[dst]
__global__ void edge_scatter(const long long* __restrict__ src,
                             const long long* __restrict__ dst,
                             const float* __restrict__ dinv,
                             const float* __restrict__ H,
                             float* __restrict__ AGG, int nedges) {
  const int lane   = threadIdx.x & 31;
  int e            = (int)((blockIdx.x * (long)blockDim.x + threadIdx.x) >> 5);
  const int nwaves = (int)(((long)gridDim.x * blockDim.x) >> 5);

  for (; e < nedges; e += nwaves) {
    const int s = (int)src[e];
    const int d = (int)dst[e];
    __builtin_prefetch(&src[e + nwaves], 0, 0);   // global_prefetch_b8
    const float norm = dinv[s] * dinv[d];

    const float4 v = ((const float4*)(H + (long)s * DFEAT))[lane];
    float* ap = AGG + (long)d * DFEAT + lane * 4;
    unsafeAtomicAdd(ap + 0, norm * v.x);
    unsafeAtomicAdd(ap + 1, norm * v.y);
    unsafeAtomicAdd(ap + 2, norm * v.z);
    unsafeAtomicAdd(ap + 3, norm * v.w);
  }
}

// ------------------------------- epilogue ----------------------------------
// AGG = [relu]( AGG + H * dinv^2 + bias )   (self-loop contribution + bias)
__global__ void epilogue(float* __restrict__ AGG, const float* __restrict__ H,
                         const float* __restrict__ dinv, const float* __restrict__ bias,
                         long total, int do_relu) {
  long i = (long)blockIdx.x * blockDim.x + threadIdx.x;
  if (i >= total) return;
  const int row = (int)(i >> 7);      // /128
  const int c   = (int)(i & 127);
  const float iv = dinv[row];
  float v = AGG[i] + H[i] * (iv * iv) + bias[c];
  if (do_relu) v = fmaxf(v, 0.0f);
  AGG[i] = v;
}

// ------------------------------- launcher ----------------------------------

extern "C" void kernel_launch(void* const* d_in, const int* in_sizes, int n_in,
                              void* d_out, int out_size, void* d_ws, size_t ws_size,
                              hipStream_t stream) {
  (void)n_in; (void)out_size; (void)ws_size;

  const float*      x  = (const float*)d_in[0];
  const long long*  ei = (const long long*)d_in[1];   // int64 per reference
  const float*      W1 = (const float*)d_in[2];
  const float*      b1 = (const float*)d_in[3];
  const float*      W2 = (const float*)d_in[4];
  const float*      b2 = (const float*)d_in[5];
  float*            out = (float*)d_out;

  const int N = in_sizes[0] / DFEAT;        // 50000
  const int E = in_sizes[1] / 2;            // 800000
  const long ND = (long)N * DFEAT;

  const long long* src = ei;                // edge_index[0]
  const long long* dst = ei + E;            // edge_index[1]

  // workspace layout (floats): dinv[N pad] | h[N*128] | agg[N*128]
  float* dinv = (float*)d_ws;
  float* h    = dinv + ((N + 255) & ~255);
  float* agg  = h + ND;

  const int T = 256;
  const int gemm_blocks = (N + 63) / 64;
  const int edge_blocks = (E + 7) / 8;      // 8 waves per block, 1 wave per edge
  const int nd_blocks   = (int)((ND + T - 1) / T);

  // ---- degree / normalization (shared by both layers) ----
  zero_f32<<<(N + T - 1) / T, T, 0, stream>>>(dinv, N);
  degree_count<<<(E + T - 1) / T, T, 0, stream>>>(dst, dinv, E);
  rsqrt_inplace<<<(N + T - 1) / T, T, 0, stream>>>(dinv, N);

  // ---- layer 1 ----
  gcn_gemm_wmma<<<gemm_blocks, T, 0, stream>>>(x, W1, h, N);
  zero_f32<<<nd_blocks, T, 0, stream>>>(agg, ND);
  edge_scatter<<<edge_blocks, T, 0, stream>>>(src, dst, dinv, h, agg, E);
  epilogue<<<nd_blocks, T, 0, stream>>>(agg, h, dinv, b1, ND, /*relu=*/1);

  // ---- layer 2 ----
  gcn_gemm_wmma<<<gemm_blocks, T, 0, stream>>>(agg, W2, h, N);
  zero_f32<<<nd_blocks, T, 0, stream>>>(out, ND);
  edge_scatter<<<edge_blocks, T, 0, stream>>>(src, dst, dinv, h, out, E);
  epilogue<<<nd_blocks, T, 0, stream>>>(out, h, dinv, b2, ND, /*relu=*/0);
}